// RCScorer_29180007809584
// MI455X (gfx1250) — compile-verified
//
#include <hip/hip_runtime.h>
#include <hip/hip_bf16.h>
#include <climits>

// ---------------------------------------------------------------------------
// RC scorer pipeline for MI455X (gfx1250, wave32)
//   h = loss @ w            (WMMA f32 16x16x4, 16 rows per wave)
//   segment-max over edges  (int-key atomic max, L2-resident h gathers)
//   softmax over node dim   (deterministic two-level reductions)
// ---------------------------------------------------------------------------

typedef __attribute__((ext_vector_type(2))) float v2f;
typedef __attribute__((ext_vector_type(8))) float v8f;

#define NBLK 256
#define NTHR 256

// Monotone int key for f32 so integer atomic max == float max.
__device__ __forceinline__ int f2key(float f) {
    int i = __float_as_int(f);
    return (i >= 0) ? i : (i ^ 0x7fffffff);
}
__device__ __forceinline__ float key2f(int k) {
    return __int_as_float((k >= 0) ? k : (k ^ 0x7fffffff));
}

// ---------------------------------------------------------------------------
// 0) init: sentinel keys for both segment-max buffers + global max scalar
// ---------------------------------------------------------------------------
__global__ void init_kernel(int* __restrict__ preK, int* __restrict__ sucK,
                            int* __restrict__ maxKey, int n) {
    int i = blockIdx.x * blockDim.x + threadIdx.x;
    if (i < n) { preK[i] = INT_MIN; sucK[i] = INT_MIN; }
    if (i == 0) *maxKey = INT_MIN;
}

// ---------------------------------------------------------------------------
// 1) h[row] = dot(loss[row,:], w)  via V_WMMA_F32_16X16X4_F32
//    One wave handles a 16-row tile; B replicates w across all 16 columns so
//    every column of D is the desired dot product; C accumulates over K.
//    A layout (ISA 7.12.2, 32-bit A 16x4): lanes 0-15 -> K={0,1},
//    lanes 16-31 -> K={2,3}; M = lane & 15.
// ---------------------------------------------------------------------------
__global__ void h_wmma_kernel(const float* __restrict__ loss,
                              const float* __restrict__ w,
                              float* __restrict__ h,
                              int n, int n_tiles, int K) {
    const int lane  = threadIdx.x & 31;
    const int wave  = threadIdx.x >> 5;
    const int tile  = blockIdx.x * (blockDim.x >> 5) + wave;
    if (tile >= n_tiles) return;                 // wave-uniform: EXEC stays all-1s

    int row = (tile << 4) + (lane & 15);
    if (row >= n) row = n - 1;                   // clamp (dup loads, stores guarded)
    const int khalf = (lane >> 4) << 1;          // 0 for lanes 0-15, 2 for 16-31
    const float* __restrict__ arow = loss + (size_t)row * (size_t)K;

    v8f c = {};
    for (int kb = 0; kb < K; kb += 4) {
        v2f a, b;
        a.x = arow[kb + khalf];
        a.y = arow[kb + khalf + 1];
        b.x = w[kb + khalf];                     // B[k][n] = w[k] for all n
        b.y = w[kb + khalf + 1];
        // (neg_a, A, neg_b, B, c_mod, C, reuse_a, reuse_b)
        c = __builtin_amdgcn_wmma_f32_16x16x4_f32(false, a, false, b,
                                                  (short)0, c, false, false);
    }

    // D layout: VGPR j, lanes 0-15 -> M=j ; lanes 16-31 -> M=8+j (all cols equal).
    if ((lane & 15) == 0) {
        const int rbase = (tile << 4) + ((lane >> 4) << 3);
#pragma unroll
        for (int j = 0; j < 8; ++j)
            if (rbase + j < n) h[rbase + j] = c[j];
    }
}

// ---------------------------------------------------------------------------
// 2) edge pass: pre[dst] = max h[src], suc[src] = max h[dst]  (skip self-loops)
// ---------------------------------------------------------------------------
__global__ void edge_kernel(const int* __restrict__ src,
                            const int* __restrict__ dst,
                            const float* __restrict__ h,
                            int* __restrict__ preK, int* __restrict__ sucK,
                            int e) {
    int i = blockIdx.x * blockDim.x + threadIdx.x;
    if (i >= e) return;
    const int s = src[i], d = dst[i];
    if (s == d) return;
    atomicMax(&preK[d], f2key(h[s]));
    atomicMax(&sucK[s], f2key(h[d]));
}

// ---------------------------------------------------------------------------
// 3) logits + global max (atomic int-key max is order-independent)
// ---------------------------------------------------------------------------
__global__ void logits_kernel(const float* __restrict__ h,
                              const int* __restrict__ preK,
                              const int* __restrict__ sucK,
                              const float* __restrict__ wg,
                              float* __restrict__ logits,
                              int* __restrict__ maxKey, int n) {
    const float g0 = wg[0], g1 = wg[1];
    int lmax = INT_MIN;
    for (int i = blockIdx.x * blockDim.x + threadIdx.x; i < n;
         i += gridDim.x * blockDim.x) {
        const int pk = preK[i], sk = sucK[i];
        const float p = (pk == INT_MIN) ? 0.0f : key2f(pk);   // DGL zero-fill
        const float s = (sk == INT_MIN) ? 0.0f : key2f(sk);
        const float L = h[i] + g0 * p + g1 * s;
        logits[i] = L;
        lmax = max(lmax, f2key(L));
    }
    __shared__ int sm[NTHR];
    sm[threadIdx.x] = lmax;
    __syncthreads();
    for (int off = NTHR >> 1; off > 0; off >>= 1) {
        if (threadIdx.x < off) sm[threadIdx.x] = max(sm[threadIdx.x], sm[threadIdx.x + off]);
        __syncthreads();
    }
    if (threadIdx.x == 0) atomicMax(maxKey, sm[0]);
}

// ---------------------------------------------------------------------------
// 4) e = exp(logit - max) -> d_out ; deterministic per-block partial sums
// ---------------------------------------------------------------------------
__global__ void exp_kernel(const float* __restrict__ logits,
                           const int* __restrict__ maxKey,
                           float* __restrict__ out,
                           float* __restrict__ partial, int n) {
    const float M = key2f(*maxKey);
    float lsum = 0.0f;
    for (int i = blockIdx.x * blockDim.x + threadIdx.x; i < n;
         i += gridDim.x * blockDim.x) {
        const float ev = expf(logits[i] - M);
        out[i] = ev;
        lsum += ev;
    }
    __shared__ float sm[NTHR];
    sm[threadIdx.x] = lsum;
    __syncthreads();
    for (int off = NTHR >> 1; off > 0; off >>= 1) {
        if (threadIdx.x < off) sm[threadIdx.x] += sm[threadIdx.x + off];
        __syncthreads();
    }
    if (threadIdx.x == 0) partial[blockIdx.x] = sm[0];
}

// ---------------------------------------------------------------------------
// 5) single-block finalize: S = sum(partial), store 1/S
// ---------------------------------------------------------------------------
__global__ void finalize_kernel(const float* __restrict__ partial,
                                float* __restrict__ invS) {
    __shared__ float sm[NBLK];
    sm[threadIdx.x] = partial[threadIdx.x];
    __syncthreads();
    for (int off = NBLK >> 1; off > 0; off >>= 1) {
        if (threadIdx.x < off) sm[threadIdx.x] += sm[threadIdx.x + off];
        __syncthreads();
    }
    if (threadIdx.x == 0) *invS = 1.0f / sm[0];
}

// ---------------------------------------------------------------------------
// 6) scale
// ---------------------------------------------------------------------------
__global__ void scale_kernel(float* __restrict__ out,
                             const float* __restrict__ invS, int n) {
    const int i = blockIdx.x * blockDim.x + threadIdx.x;
    if (i < n) out[i] *= *invS;
}

// ---------------------------------------------------------------------------
extern "C" void kernel_launch(void* const* d_in, const int* in_sizes, int n_in,
                              void* d_out, int out_size, void* d_ws, size_t ws_size,
                              hipStream_t stream) {
    const float* loss = (const float*)d_in[0];   // [N, K]
    const float* wser = (const float*)d_in[1];   // [1, K]
    const float* wgra = (const float*)d_in[2];   // [1, 2]
    const int*   esrc = (const int*)d_in[3];     // [E]
    const int*   edst = (const int*)d_in[4];     // [E]

    const int K = in_sizes[1];                   // 200
    const int n = in_sizes[0] / K;               // 100000
    const int e = in_sizes[3];                   // 3200000

    // Workspace layout
    char* ws = (char*)d_ws;
    float* h       = (float*)(ws);
    int*   preK    = (int*)  (ws + (size_t)n * 4);
    int*   sucK    = (int*)  (ws + (size_t)n * 8);
    float* logits  = (float*)(ws + (size_t)n * 12);
    float* partial = (float*)(ws + (size_t)n * 16);
    int*   maxKey  = (int*)  (ws + (size_t)n * 16 + NBLK * 4);
    float* invS    = (float*)(ws + (size_t)n * 16 + NBLK * 4 + 4);

    float* out = (float*)d_out;

    const int gN = (n + NTHR - 1) / NTHR;
    const int gE = (e + NTHR - 1) / NTHR;
    const int n_tiles = (n + 15) / 16;
    const int waves_per_block = NTHR / 32;       // 8 tiles per block
    const int gH = (n_tiles + waves_per_block - 1) / waves_per_block;

    init_kernel<<<gN, NTHR, 0, stream>>>(preK, sucK, maxKey, n);
    h_wmma_kernel<<<gH, NTHR, 0, stream>>>(loss, wser, h, n, n_tiles, K);
    edge_kernel<<<gE, NTHR, 0, stream>>>(esrc, edst, h, preK, sucK, e);
    logits_kernel<<<NBLK, NTHR, 0, stream>>>(h, preK, sucK, wgra, logits, maxKey, n);
    exp_kernel<<<NBLK, NTHR, 0, stream>>>(logits, maxKey, out, partial, n);
    finalize_kernel<<<1, NBLK, 0, stream>>>(partial, invS);
    scale_kernel<<<gN, NTHR, 0, stream>>>(out, invS, n);
}